// Model_2894807958097
// MI455X (gfx1250) — compile-verified
//
#include <hip/hip_runtime.h>

typedef float v4f __attribute__((ext_vector_type(4)));

namespace {
constexpr int   kB        = 64;
constexpr int   kOH       = 120;
constexpr int   kOW       = 160;
constexpr int   kHW       = kOH * kOW;     // 19200
constexpr int   kNV       = kHW / 4;       // 4800 vec4 groups per batch image
constexpr int   kTotalV   = kB * kNV;      // 307200 threads
constexpr float kInH      = 480.0f;        // out_h * 4
constexpr float kInW      = 640.0f;        // out_w * 4
constexpr float kThresh   = 0.5f;
constexpr int   kBoxElems = kB * kHW * 5;  // 6,144,000
constexpr int   kLmElems  = kB * kHW * 10; // 12,288,000
} // namespace

__global__ __launch_bounds__(256) void centerface_decode_kernel(
    const float* __restrict__ heat,
    const float* __restrict__ scale,
    const float* __restrict__ offset,
    const float* __restrict__ lmk,
    float* __restrict__ boxes_out,
    float* __restrict__ lms_out,
    float* __restrict__ keep_out)
{
  const int g = blockIdx.x * 256 + threadIdx.x;
  if (g >= kTotalV) return;

  const int b = g / kNV;          // batch image
  const int v = g - b * kNV;      // vec4 group within image
  const int p = v * 4;            // first spatial position (never crosses a row)
  const float rowy = (float)(v / 40);  // p / 160: shared row index for all 4 positions

  // ---- Streaming b128 loads (each input byte read exactly once -> NT hint) ----
  const v4f h   = __builtin_nontemporal_load((const v4f*)(heat   + (size_t)b * kHW + p));
  const v4f sc0 = __builtin_nontemporal_load((const v4f*)(scale  + ((size_t)b * 2 + 0) * kHW + p));
  const v4f sc1 = __builtin_nontemporal_load((const v4f*)(scale  + ((size_t)b * 2 + 1) * kHW + p));
  const v4f o0  = __builtin_nontemporal_load((const v4f*)(offset + ((size_t)b * 2 + 0) * kHW + p));
  // NOTE: offset channel 1 is dead in the reference (o1 computed but never used) -> skip it.
  v4f lm[10];
#pragma unroll
  for (int c = 0; c < 10; ++c)
    lm[c] = __builtin_nontemporal_load((const v4f*)(lmk + ((size_t)b * 10 + c) * kHW + p));

  float bx[20];   // 4 positions x [x1, y1, x2, y2, s]
  float lo[40];   // 4 positions x 10 interleaved [y0,x0,y1,x1,...]
  v4f   kv;       // keep flags as 1.0 / 0.0

#pragma unroll
  for (int k = 0; k < 4; ++k) {
    const float e0 = __expf(sc0[k]) * 4.0f;                       // s0 = exp(scale0)*4
    const float e1 = __expf(sc1[k]) * 4.0f;                       // s1 = exp(scale1)*4
    const float y1raw = (rowy + o0[k] + 0.5f) * 4.0f - e0 * 0.5f;
    const float y1c = fmaxf(y1raw, 0.0f);
    const float y1  = fminf(y1c, kInH);
    const float x1  = fminf(y1c, kInW);   // reference quirk: x1 from the y-clamped value
    const float y2  = fminf(y1 + e0, kInH);
    const float x2  = fminf(x1 + e1, kInW);
    const float sv  = h[k];
    const bool  kp  = sv >= kThresh;

    bx[5 * k + 0] = kp ? x1 : 0.0f;
    bx[5 * k + 1] = kp ? y1 : 0.0f;
    bx[5 * k + 2] = kp ? x2 : 0.0f;
    bx[5 * k + 3] = kp ? y2 : 0.0f;
    bx[5 * k + 4] = kp ? sv : 0.0f;

#pragma unroll
    for (int j = 0; j < 5; ++j) {
      // out[2j]   = lm_even * s0 + y1 ; out[2j+1] = lm_odd * s1 + x1
      lo[10 * k + 2 * j + 0] = kp ? (lm[2 * j + 0][k] * e0 + y1) : 0.0f;
      lo[10 * k + 2 * j + 1] = kp ? (lm[2 * j + 1][k] * e1 + x1) : 0.0f;
    }
    kv[k] = kp ? 1.0f : 0.0f;
  }

  // ---- Streaming b128 stores: per-thread regions are contiguous & 16B aligned ----
  v4f* bo = (v4f*)(boxes_out + (size_t)b * kHW * 5 + (size_t)p * 5);   // 80 B / thread
#pragma unroll
  for (int i = 0; i < 5; ++i) {
    v4f t = { bx[4 * i + 0], bx[4 * i + 1], bx[4 * i + 2], bx[4 * i + 3] };
    __builtin_nontemporal_store(t, bo + i);
  }

  v4f* lmo = (v4f*)(lms_out + (size_t)b * kHW * 10 + (size_t)p * 10);  // 160 B / thread
#pragma unroll
  for (int i = 0; i < 10; ++i) {
    v4f t = { lo[4 * i + 0], lo[4 * i + 1], lo[4 * i + 2], lo[4 * i + 3] };
    __builtin_nontemporal_store(t, lmo + i);
  }

  __builtin_nontemporal_store(kv, (v4f*)(keep_out + (size_t)b * kHW + p));
}

extern "C" void kernel_launch(void* const* d_in, const int* in_sizes, int n_in,
                              void* d_out, int out_size, void* d_ws, size_t ws_size,
                              hipStream_t stream) {
  const float* heat   = (const float*)d_in[0];  // (64, 1, 120, 160)
  const float* scale  = (const float*)d_in[1];  // (64, 2, 120, 160)
  const float* offset = (const float*)d_in[2];  // (64, 2, 120, 160)
  const float* lmk    = (const float*)d_in[3];  // (64, 10, 120, 160)

  float* out   = (float*)d_out;                 // tuple concat: boxes | landmarks | keep
  float* boxes = out;
  float* lms   = out + kBoxElems;
  float* keep  = out + kBoxElems + kLmElems;

  const int blocks = (kTotalV + 255) / 256;     // 1200 blocks x 256 threads (8 wave32/block)
  hipLaunchKernelGGL(centerface_decode_kernel, dim3(blocks), dim3(256), 0, stream,
                     heat, scale, offset, lmk, boxes, lms, keep);
}